// MAMGCN_encblock_54941221651386
// MI455X (gfx1250) — compile-verified
//
#include <hip/hip_runtime.h>
#include <math.h>

// ---------------------------------------------------------------------------
// Problem constants (from the reference)
// ---------------------------------------------------------------------------
constexpr int Bb   = 8;
constexpr int Nn   = 307;
constexpr int Tt   = 12;
constexpr int Ff   = 64;
constexpr int ED   = 128;
constexpr int DSs  = 16;
constexpr int BN   = Bb * Nn;          // 2456
constexpr int R    = BN * Tt;          // 29472 rows (== B*T*N as well)
constexpr int NEL  = R * Ff;           // 1886208 elements per (B,N,T,F) tensor
constexpr int NN_  = Nn * Nn;          // 94249
constexpr int BNN  = Bb * NN_;         // 753992
constexpr int PANEL = Nn * Tt * Ff;    // 235776 (one batch's trend panel)

typedef __attribute__((ext_vector_type(2))) float v2f;
typedef __attribute__((ext_vector_type(8))) float v8f;

// ---------------------------------------------------------------------------
// fp32 WMMA 16x16x4 (D = A(16x4) * B(4x16) + C) -- CDNA5 matrix pipe, fp32
// exact semantics wrt the fp32 reference.  Layout per ISA 7.12.2:
//   A: lanes 0-15 hold row M=lane, K=0..1 in vgpr0/1; lanes 16-31 K=2..3.
//   B: assumed symmetric (lanes 0-15: col N=lane of rows K=0,1; hi: K=2,3).
//   C/D: vgpr r holds row (r + 8*hiHalf), col = lane&15.
// ---------------------------------------------------------------------------
__device__ inline v8f wmma4_f32(v2f a, v2f b, v8f c) {
#if __has_builtin(__builtin_amdgcn_wmma_f32_16x16x4_f32)
  return __builtin_amdgcn_wmma_f32_16x16x4_f32(false, a, false, b, (short)0, c,
                                               false, false);
#else
  // Shuffle-based emulation (compile-robust fallback; same math).
  int lane = threadIdx.x & 31;
  int half = lane >> 4, lm = lane & 15;
  for (int r = 0; r < 8; ++r) {
    int m = half * 8 + r;
    float sum = 0.f;
    for (int k = 0; k < 4; ++k) {
      float av = __shfl((k & 1) ? a.y : a.x, (k >= 2 ? 16 : 0) + m, 32);
      float bv = __shfl((k & 1) ? b.y : b.x, (k >= 2 ? 16 : 0) + lm, 32);
      sum += av * bv;
    }
    c[r] += sum;
  }
  return c;
#endif
}

__device__ inline float epi_act(float v, int act) {
  if (act == 1) return v > 0.f ? v : 0.f;
  if (act == 2) return 1.f / (1.f + expf(-v));
  return v;
}

// ---------------------------------------------------------------------------
// Strip WMMA GEMM: one wave computes a 16x64 strip of C (4 accumulators).
// Requires M%16==0, N%64==0, K%4==0 -> fully unguarded: per K-step the A
// fragment is loaded once and 4 back-to-back v_wmma consume it against 4
// B tiles (10 loads / 8192 FLOP).  Covers all bulk (M=29472) GEMMs here except N=36.
// ---------------------------------------------------------------------------
__global__ __launch_bounds__(32) void gemm_wmma_strip_kernel(
    const float* __restrict__ A, const float* __restrict__ W,
    const float* __restrict__ bias, const float* __restrict__ res,
    float* __restrict__ C, int Kd, int lda, int ldb, int ldc, int ldres,
    int accum, int act) {
  int lane = threadIdx.x & 31;
  int half = lane >> 4;
  int lm = lane & 15;
  int row = blockIdx.y * 16 + lm;
  int col0 = blockIdx.x * 64 + lm;

  v8f acc0 = {0.f, 0.f, 0.f, 0.f, 0.f, 0.f, 0.f, 0.f};
  v8f acc1 = acc0, acc2 = acc0, acc3 = acc0;
  if (accum) {
#pragma unroll
    for (int r = 0; r < 8; ++r) {
      long long m = blockIdx.y * 16 + half * 8 + r;
      acc0[r] = C[m * ldc + col0];
      acc1[r] = C[m * ldc + col0 + 16];
      acc2[r] = C[m * ldc + col0 + 32];
      acc3[r] = C[m * ldc + col0 + 48];
    }
  }

  const float* ap = A + (long long)row * lda + half * 2;
  const float* bp = W + (long long)(half * 2) * ldb + col0;
  long long bstep = 4LL * (long long)ldb;
  for (int kk = 0; kk < Kd; kk += 4) {
    v2f a;
    a.x = ap[0];
    a.y = ap[1];
    ap += 4;
    v2f b0, b1, b2, b3;
    b0.x = bp[0];
    b0.y = bp[ldb];
    b1.x = bp[16];
    b1.y = bp[ldb + 16];
    b2.x = bp[32];
    b2.y = bp[ldb + 32];
    b3.x = bp[48];
    b3.y = bp[ldb + 48];
    bp += bstep;
    acc0 = wmma4_f32(a, b0, acc0);
    acc1 = wmma4_f32(a, b1, acc1);
    acc2 = wmma4_f32(a, b2, acc2);
    acc3 = wmma4_f32(a, b3, acc3);
  }

#pragma unroll
  for (int r = 0; r < 8; ++r) {
    long long m = blockIdx.y * 16 + half * 8 + r;
#pragma unroll
    for (int j = 0; j < 4; ++j) {
      int c = col0 + 16 * j;
      float v = (j == 0) ? acc0[r] : (j == 1) ? acc1[r] : (j == 2) ? acc2[r]
                                                                   : acc3[r];
      if (bias) v += bias[c];
      if (res) v += res[m * ldres + c];
      C[m * ldc + c] = epi_act(v, act);
    }
  }
}

// ---------------------------------------------------------------------------
// Generic batched WMMA GEMM (edge shapes: N=307 / K=307 / N=36 / batched).
// One wave = one 16x16 tile; interior fast path, select-guarded edge path.
// act: 0 none, 1 relu, 2 sigmoid.  bMod!=0 -> W batch index = z % bMod.
// ---------------------------------------------------------------------------
__global__ __launch_bounds__(32) void gemm_wmma_kernel(
    const float* __restrict__ A, const float* __restrict__ W,
    const float* __restrict__ bias, const float* __restrict__ res,
    float* __restrict__ C, int M, int N, int Kd, int lda, int ldb, int ldc,
    int ldres, long long sA, long long sB, long long sC, int bMod, int accum,
    int act) {
  int z = blockIdx.z;
  A += (long long)z * sA;
  W += (long long)(bMod ? (z % bMod) : z) * sB;
  C += (long long)z * sC;

  int lane = threadIdx.x & 31;
  int half = lane >> 4;
  int lm = lane & 15;
  int row0 = blockIdx.y * 16;
  int col0 = blockIdx.x * 16;
  int row = row0 + lm;  // A row for this lane (both halves)
  int col = col0 + lm;  // C/W column for this lane

  v8f acc = {0.f, 0.f, 0.f, 0.f, 0.f, 0.f, 0.f, 0.f};
  if (accum) {
    for (int r = 0; r < 8; ++r) {
      int m = row0 + half * 8 + r;
      acc[r] = (m < M && col < N) ? C[(long long)m * ldc + col] : 0.f;
    }
  }

  bool interior = (row0 + 16 <= M) && (col0 + 16 <= N) && ((Kd & 3) == 0);
  if (interior) {
    const float* ap = A + (long long)row * lda + half * 2;
    const float* bp = W + (long long)(half * 2) * ldb + col;
    long long bstep = 4LL * (long long)ldb;
    for (int kk = 0; kk < Kd; kk += 4) {
      v2f a, b;
      a.x = ap[0];
      a.y = ap[1];
      b.x = bp[0];
      b.y = bp[ldb];
      ap += 4;
      bp += bstep;
      acc = wmma4_f32(a, b, acc);
    }
  } else {
    bool rOK = row < M, cOK = col < N;
    const float* ap = A + (long long)(rOK ? row : 0) * lda;
    int cc = cOK ? col : 0;
    for (int kk = 0; kk < Kd; kk += 4) {
      int k0 = kk + half * 2;
      bool k0ok = k0 < Kd, k1ok = (k0 + 1) < Kd;
      int k0c = k0ok ? k0 : 0;
      int k1c = k1ok ? k0 + 1 : 0;
      float a0 = ap[k0c], a1 = ap[k1c];
      float b0 = W[(long long)k0c * ldb + cc];
      float b1 = W[(long long)k1c * ldb + cc];
      v2f a, b;
      a.x = (rOK && k0ok) ? a0 : 0.f;
      a.y = (rOK && k1ok) ? a1 : 0.f;
      b.x = (cOK && k0ok) ? b0 : 0.f;
      b.y = (cOK && k1ok) ? b1 : 0.f;
      acc = wmma4_f32(a, b, acc);
    }
  }

  for (int r = 0; r < 8; ++r) {
    int m = row0 + half * 8 + r;
    if (m < M && col < N) {
      float v = acc[r];
      if (bias) v += bias[col];
      if (res) v += res[(long long)m * ldres + col];
      C[(long long)m * ldc + col] = epi_act(v, act);
    }
  }
}

// ---------------------------------------------------------------------------
// Elementwise / small kernels
// ---------------------------------------------------------------------------
__device__ inline float sigm(float x) { return 1.f / (1.f + expf(-x)); }
__device__ inline float softplus_(float x) {
  return x > 20.f ? x : log1pf(expf(x));
}

// Series decomposition (multi moving-avg + softmax mix) with STE added.
__global__ void k_decomp(const float* __restrict__ x, const float* __restrict__ TE,
                         const float* __restrict__ SE, const float* __restrict__ dw,
                         const float* __restrict__ db, float* __restrict__ trend,
                         float* __restrict__ seas) {
  int i = blockIdx.x * blockDim.x + threadIdx.x;
  if (i >= NEL) return;
  int f = i & 63;
  int t = (i >> 6) % Tt;
  int n = (i / (Ff * Tt)) % Nn;
  int b = i / (Ff * Tt * Nn);
  const float* row = x + (size_t)(i / (Ff * Tt)) * (Ff * Tt);
  float xv = x[i];
  float s3 = 0.f;
  for (int d = -1; d <= 1; ++d) {
    int tt = t + d; tt = tt < 0 ? 0 : (tt > Tt - 1 ? Tt - 1 : tt);
    s3 += row[tt * Ff + f];
  }
  s3 *= (1.f / 3.f);
  float s7 = 0.f;
  for (int d = -3; d <= 3; ++d) {
    int tt = t + d; tt = tt < 0 ? 0 : (tt > Tt - 1 ? Tt - 1 : tt);
    s7 += row[tt * Ff + f];
  }
  s7 *= (1.f / 7.f);
  float a0 = xv * dw[0] + db[0], a1 = xv * dw[1] + db[1];
  float mx = fmaxf(a0, a1);
  float e0 = expf(a0 - mx), e1 = expf(a1 - mx);
  float w0 = e0 / (e0 + e1);
  float mean = s3 * w0 + s7 * (1.f - w0);
  float ste = TE[(b * Tt + t) * Ff + f] + SE[n * Ff + f];
  trend[i] = mean + ste;
  seas[i] = (xv - mean) + ste;
}

// lhs[b,n,t] = sum_f (sum_tt trend[b,n,tt,f]*W1[tt]) * W2[f,t]
// rhs[b,t,n] = sum_f W3[f]*trend[b,n,t,f]
__global__ void k_sat_lr(const float* __restrict__ trend, const float* __restrict__ W1,
                         const float* __restrict__ W2, const float* __restrict__ W3,
                         float* __restrict__ lhs, float* __restrict__ rhs) {
  int i = blockIdx.x * blockDim.x + threadIdx.x;
  if (i >= Bb * Nn * Tt) return;
  int t = i % Tt;
  int n = (i / Tt) % Nn;
  int b = i / (Tt * Nn);
  const float* row = trend + (size_t)(b * Nn + n) * Tt * Ff;
  float acc = 0.f;
  for (int f = 0; f < Ff; ++f) {
    float u = 0.f;
    for (int tt = 0; tt < Tt; ++tt) u += row[tt * Ff + f] * W1[tt];
    acc += u * W2[f * Tt + t];
  }
  lhs[i] = acc;
  float r = 0.f;
  for (int f = 0; f < Ff; ++f) r += W3[f] * row[t * Ff + f];
  rhs[(b * Tt + t) * Nn + n] = r;
}

// P[b,m,l] = sigmoid(sum_t lhs[b,m,t]*rhs[b,t,l] + bs[m,l])
__global__ void k_prod_sig(const float* __restrict__ lhs, const float* __restrict__ rhs,
                           const float* __restrict__ bs, float* __restrict__ P) {
  int i = blockIdx.x * blockDim.x + threadIdx.x;
  if (i >= BNN) return;
  int l = i % Nn;
  int m = (i / Nn) % Nn;
  int b = i / NN_;
  float acc = bs[m * Nn + l];
  for (int t = 0; t < Tt; ++t)
    acc += lhs[(b * Nn + m) * Tt + t] * rhs[(b * Tt + t) * Nn + l];
  P[i] = sigm(acc);
}

// softmax over dim 1 (n) of S[b,n,l]
__global__ void k_softmax_n(const float* __restrict__ S, float* __restrict__ out) {
  int i = blockIdx.x * blockDim.x + threadIdx.x;
  if (i >= Bb * Nn) return;
  int l = i % Nn;
  int b = i / Nn;
  float mx = -3.0e38f;
  for (int n = 0; n < Nn; ++n) mx = fmaxf(mx, S[(size_t)(b * Nn + n) * Nn + l]);
  float sum = 0.f;
  for (int n = 0; n < Nn; ++n) sum += expf(S[(size_t)(b * Nn + n) * Nn + l] - mx);
  float inv = 1.f / sum;
  for (int n = 0; n < Nn; ++n) {
    size_t idx = (size_t)(b * Nn + n) * Nn + l;
    out[idx] = expf(S[idx] - mx) * inv;
  }
}

// TkT[(k*B+b)][n][m] = cheb[k,m,n] * SAt[b,m,n]   (transposed for GEMM)
__global__ void k_tkt(const float* __restrict__ cheb, const float* __restrict__ sat,
                      float* __restrict__ TkT) {
  int i = blockIdx.x * blockDim.x + threadIdx.x;
  if (i >= 3 * BNN) return;
  int m = i % Nn;
  int n = (i / Nn) % Nn;
  int z = i / NN_;
  int k = z / Bb, b = z % Bb;
  TkT[i] = cheb[(size_t)(k * Nn + m) * Nn + n] * sat[(size_t)(b * Nn + m) * Nn + n];
}

__global__ void k_rmsnorm(const float* __restrict__ in, const float* __restrict__ w,
                          float* __restrict__ out) {
  int r = blockIdx.x * blockDim.x + threadIdx.x;
  if (r >= R) return;
  const float* a = in + (size_t)r * Ff;
  float ss = 0.f;
  for (int f = 0; f < Ff; ++f) ss += a[f] * a[f];
  float inv = rsqrtf(ss * (1.f / Ff) + 1e-5f);
  float* o = out + (size_t)r * Ff;
  for (int f = 0; f < Ff; ++f) o[f] = a[f] * inv * w[f];
}

// Two-pass LayerNorm over F=64 (avoids a 64-float register array / spills).
__global__ void k_layernorm(const float* __restrict__ a, const float* __restrict__ res,
                            const float* __restrict__ g, const float* __restrict__ bb,
                            float* __restrict__ out) {
  int r = blockIdx.x * blockDim.x + threadIdx.x;
  if (r >= R) return;
  const float* pa = a + (size_t)r * Ff;
  const float* pr = res ? res + (size_t)r * Ff : nullptr;
  float mu = 0.f, m2 = 0.f;
  for (int f = 0; f < Ff; ++f) {
    float t = pa[f] + (pr ? pr[f] : 0.f);
    mu += t;
    m2 += t * t;
  }
  mu *= (1.f / Ff);
  float var = m2 * (1.f / Ff) - mu * mu;
  var = var > 0.f ? var : 0.f;
  float inv = rsqrtf(var + 1e-5f);
  float* po = out + (size_t)r * Ff;
  for (int f = 0; f < Ff; ++f) {
    float t = pa[f] + (pr ? pr[f] : 0.f);
    po[f] = (t - mu) * inv * g[f] + bb[f];
  }
}

// causal depthwise conv (kernel 4, left zero-pad 3) + bias + SiLU
__global__ void k_conv_silu(const float* __restrict__ inproj, const float* __restrict__ w,
                            const float* __restrict__ b, float* __restrict__ out) {
  int i = blockIdx.x * blockDim.x + threadIdx.x;
  if (i >= R * ED) return;
  int c = i % ED;
  int r = i / ED;
  int t = r % Tt;
  int bn = r / Tt;
  float acc = b[c];
  for (int j = 0; j < 4; ++j) {
    int tt = t - 3 + j;
    if (tt >= 0) acc += w[c * 4 + j] * inproj[(size_t)(bn * Tt + tt) * 256 + c];
  }
  out[i] = acc * sigm(acc);
}

// Selective scan fused with delta-projection/softplus and u = y*silu(z).
__global__ void k_scan(const float* __restrict__ dbc, const float* __restrict__ dtw,
                       const float* __restrict__ dtb, const float* __restrict__ Alog,
                       const float* __restrict__ Dp, const float* __restrict__ inproj,
                       float* __restrict__ xr) {
  int i = blockIdx.x * blockDim.x + threadIdx.x;
  if (i >= BN * ED) return;
  int c = i % ED;
  int bn = i / ED;
  float Av[DSs];
  for (int s = 0; s < DSs; ++s) Av[s] = -expf(Alog[c * DSs + s]);
  float Dv = Dp[c];
  float dtbv = dtb[c];
  float h[DSs];
  for (int s = 0; s < DSs; ++s) h[s] = 0.f;
  for (int t = 0; t < Tt; ++t) {
    int row = bn * Tt + t;
    const float* d36 = dbc + (size_t)row * 36;
    float del = dtbv;
    for (int j = 0; j < 4; ++j) del += d36[j] * dtw[j * ED + c];
    del = softplus_(del);
    float xrv = xr[(size_t)row * ED + c];
    float acc = 0.f;
    for (int s = 0; s < DSs; ++s) {
      float dA = expf(del * Av[s]);
      h[s] = dA * h[s] + del * d36[4 + s] * xrv;
      acc += h[s] * d36[20 + s];
    }
    float y = acc + Dv * xrv;
    float z = inproj[(size_t)row * 256 + ED + c];
    xr[(size_t)row * ED + c] = y * z * sigm(z);  // in-place: xr read above
  }
}

__global__ void k_flip(const float* __restrict__ in, float* __restrict__ out, int cols) {
  int i = blockIdx.x * blockDim.x + threadIdx.x;
  int total = BN * Tt * cols;
  if (i >= total) return;
  int c = i % cols;
  int t = (i / cols) % Tt;
  int bn = i / (cols * Tt);
  out[i] = in[(size_t)(bn * Tt + (Tt - 1 - t)) * cols + c];
}

__global__ void k_flip_to_cat(const float* __restrict__ in, float* __restrict__ cat) {
  int i = blockIdx.x * blockDim.x + threadIdx.x;
  if (i >= NEL) return;
  int f = i % Ff;
  int t = (i / Ff) % Tt;
  int bn = i / (Ff * Tt);
  cat[(size_t)(bn * Tt + t) * 128 + 64 + f] =
      in[(size_t)(bn * Tt + (Tt - 1 - t)) * Ff + f];
}

__global__ void k_concat2(const float* __restrict__ a, const float* __restrict__ b,
                          float* __restrict__ cat) {
  int i = blockIdx.x * blockDim.x + threadIdx.x;
  if (i >= NEL) return;
  int f = i % Ff;
  size_t r = i / Ff;
  cat[r * 128 + f] = a[i];
  cat[r * 128 + 64 + f] = b[i];
}

// Online-softmax MHA (dh=8).  mask (adj) optional.
__global__ void k_attn(const float* __restrict__ qkv, const int* __restrict__ mask,
                       float* __restrict__ out, int batches, int seq) {
  int i = blockIdx.x * blockDim.x + threadIdx.x;
  int total = batches * 8 * seq;
  if (i >= total) return;
  int q = i % seq;
  int h = (i / seq) & 7;
  int b = i / (seq * 8);
  const float* qp = qkv + (size_t)(b * seq + q) * 192 + h * 8;
  float qv[8];
  for (int d = 0; d < 8; ++d) qv[d] = qp[d];
  float m = -3.0e38f, l = 0.f, acc[8];
  for (int d = 0; d < 8; ++d) acc[d] = 0.f;
  for (int kk = 0; kk < seq; ++kk) {
    const float* kp = qkv + (size_t)(b * seq + kk) * 192 + 64 + h * 8;
    float s = 0.f;
    for (int d = 0; d < 8; ++d) s += qv[d] * kp[d];
    s *= 0.35355339059327373f;  // 1/sqrt(8)
    if (mask && mask[(size_t)q * seq + kk] <= 0) s = -1.0e9f;
    float mn = fmaxf(m, s);
    float corr = expf(m - mn);
    float p = expf(s - mn);
    l = l * corr + p;
    const float* vp = kp + 64;
    for (int d = 0; d < 8; ++d) acc[d] = acc[d] * corr + p * vp[d];
    m = mn;
  }
  float inv = 1.f / l;
  float* op = out + (size_t)(b * seq + q) * Ff + h * 8;
  for (int d = 0; d < 8; ++d) op[d] = acc[d] * inv;
}

// (B,N,T,F) -> (B,T,N,F)
__global__ void k_to_btnf(const float* __restrict__ in, float* __restrict__ out) {
  int i = blockIdx.x * blockDim.x + threadIdx.x;
  if (i >= NEL) return;
  int f = i % Ff;
  int n = (i / Ff) % Nn;
  int t = (i / (Ff * Nn)) % Tt;
  int b = i / (Ff * Nn * Tt);
  out[i] = in[(size_t)((b * Nn + n) * Tt + t) * Ff + f];
}
// (B,T,N,F) -> (B,N,T,F)
__global__ void k_from_btnf(const float* __restrict__ in, float* __restrict__ out) {
  int i = blockIdx.x * blockDim.x + threadIdx.x;
  if (i >= NEL) return;
  int f = i % Ff;
  int t = (i / Ff) % Tt;
  int n = (i / (Ff * Tt)) % Nn;
  int b = i / (Ff * Tt * Nn);
  out[i] = in[(size_t)((b * Tt + t) * Nn + n) * Ff + f];
}

__global__ void k_blend(const float* __restrict__ g, const float* __restrict__ a,
                        const float* __restrict__ b, float* __restrict__ out) {
  int i = blockIdx.x * blockDim.x + threadIdx.x;
  if (i >= NEL) return;
  float gv = g[i];
  out[i] = gv * a[i] + (1.f - gv) * b[i];
}

__global__ void k_final(const float* __restrict__ x, const float* __restrict__ g,
                        const float* __restrict__ ses, const float* __restrict__ set,
                        float* __restrict__ out) {
  int i = blockIdx.x * blockDim.x + threadIdx.x;
  if (i >= NEL) return;
  float gv = g[i];
  out[i] = x[i] + out[i] + gv * ses[i] + (1.f - gv) * set[i];
}

// ---------------------------------------------------------------------------
// Host side
// ---------------------------------------------------------------------------
static void gemm(hipStream_t st, const float* A, const float* W, const float* bias,
                 const float* res, float* C, int M, int N, int K, int lda, int ldb,
                 int ldc, int ldres, int batch, long long sA, long long sB,
                 long long sC, int bMod, int accum, int act) {
  if (batch == 1 && (M % 16) == 0 && (N % 64) == 0 && (K % 4) == 0) {
    dim3 grid(N / 64, M / 16, 1);
    gemm_wmma_strip_kernel<<<grid, dim3(32), 0, st>>>(A, W, bias, res, C, K, lda,
                                                      ldb, ldc, ldres, accum, act);
  } else {
    dim3 grid((N + 15) / 16, (M + 15) / 16, batch);
    gemm_wmma_kernel<<<grid, dim3(32), 0, st>>>(A, W, bias, res, C, M, N, K, lda,
                                                ldb, ldc, ldres, sA, sB, sC, bMod,
                                                accum, act);
  }
}

static inline dim3 gr(long long n) { return dim3((unsigned)((n + 255) / 256)); }

extern "C" void kernel_launch(void* const* d_in, const int* in_sizes, int n_in,
                              void* d_out, int out_size, void* d_ws, size_t ws_size,
                              hipStream_t stream) {
  (void)in_sizes; (void)n_in; (void)out_size; (void)ws_size;
  const float* x     = (const float*)d_in[0];
  const float* TE    = (const float*)d_in[1];
  const float* SE    = (const float*)d_in[2];
  const int*   adj   = (const int*)d_in[3];
  const float* chebP = (const float*)d_in[4];
  const float* chebT = (const float*)d_in[5];
  const float* dec_w = (const float*)d_in[6];
  const float* dec_b = (const float*)d_in[7];
  const float* satW1 = (const float*)d_in[8];
  const float* satW2 = (const float*)d_in[9];
  const float* satW3 = (const float*)d_in[10];
  const float* satBs = (const float*)d_in[11];
  const float* satVs = (const float*)d_in[12];
  const float* mNormW  = (const float*)d_in[13];
  const float* mInProj = (const float*)d_in[14];
  const float* mConvW  = (const float*)d_in[15];
  const float* mConvB  = (const float*)d_in[16];
  const float* mXProj  = (const float*)d_in[17];
  const float* mDtW    = (const float*)d_in[18];
  const float* mDtB    = (const float*)d_in[19];
  const float* mAlog   = (const float*)d_in[20];
  const float* mD      = (const float*)d_in[21];
  const float* mOutP   = (const float*)d_in[22];
  const float* bimW  = (const float*)d_in[23];
  const float* bimB  = (const float*)d_in[24];
  const float* bimG  = (const float*)d_in[25];
  const float* bimBe = (const float*)d_in[26];
  const float* gQkvW = (const float*)d_in[27];
  const float* gQkvB = (const float*)d_in[28];
  const float* gOutW = (const float*)d_in[29];
  const float* gOutB = (const float*)d_in[30];
  const float* gLn1g = (const float*)d_in[31];
  const float* gLn1b = (const float*)d_in[32];
  const float* gFf1w = (const float*)d_in[33];
  const float* gFf1b = (const float*)d_in[34];
  const float* gFf2w = (const float*)d_in[35];
  const float* gFf2b = (const float*)d_in[36];
  const float* gLn2g = (const float*)d_in[37];
  const float* gLn2b = (const float*)d_in[38];
  const float* tQkvW = (const float*)d_in[39];
  const float* tQkvB = (const float*)d_in[40];
  const float* tOutW = (const float*)d_in[41];
  const float* tOutB = (const float*)d_in[42];
  const float* tLn1g = (const float*)d_in[43];
  const float* tLn1b = (const float*)d_in[44];
  const float* tFf1w = (const float*)d_in[45];
  const float* tFf1b = (const float*)d_in[46];
  const float* tFf2w = (const float*)d_in[47];
  const float* tFf2b = (const float*)d_in[48];
  const float* tLn2g = (const float*)d_in[49];
  const float* tLn2b = (const float*)d_in[50];
  const float* tgW = (const float*)d_in[51];
  const float* tgB = (const float*)d_in[52];
  const float* sgW = (const float*)d_in[53];
  const float* sgB = (const float*)d_in[54];
  float* out = (float*)d_out;

  // ---- workspace layout (floats), lifetime-based aliasing --------------
  float* w = (float*)d_ws;
  size_t o = 0;
  auto take = [&](size_t nf) { float* p = w + o; o += nf; return p; };
  float* trend = take(NEL);
  float* seas  = take(NEL);
  float* tes   = take(NEL);   // trend_enc_s
  float* tet   = take(NEL);   // trend_enc_t
  float* sesB  = take(NEL);   // seasonal_enc_s
  float* setB  = take(NEL);   // seasonal_enc_t
  float* gen1  = take(NEL);
  float* gen2  = take(NEL);
  float* gbuf  = take(NEL);
  float* cat   = take((size_t)R * 128);
  float* big   = take((size_t)R * 256);
  float* xrb   = take((size_t)R * 128);
  float* qkvZ  = take((size_t)R * 192);  // == 3*NEL
  // aliases (non-overlapping lifetimes)
  float* lhs  = cat;                     // during SAt
  float* rhs  = cat + (size_t)Bb * Nn * Tt;
  float* Pbuf = xrb;                     // sigmoid(prod+bs), then SAt
  float* Sbuf = xrb + BNN;
  float* SAt  = Pbuf;
  float* TkT  = big;                     // before mamba uses big
  float* Zb   = qkvZ;                    // cheb stage-1 output
  float* dbc  = qkvZ;                    // during mamba (Z dead)

  const int TPB = 256;

  // 1) decomposition + STE
  k_decomp<<<gr(NEL), TPB, 0, stream>>>(x, TE, SE, dec_w, dec_b, trend, seas);

  // 2) spatial attention S-matrix
  k_sat_lr<<<gr((long long)Bb * Nn * Tt), TPB, 0, stream>>>(trend, satW1, satW2,
                                                            satW3, lhs, rhs);
  k_prod_sig<<<gr(BNN), TPB, 0, stream>>>(lhs, rhs, satBs, Pbuf);
  // S[b] = Vs @ P[b]   (batched over b, Vs shared)
  gemm(stream, satVs, Pbuf, nullptr, nullptr, Sbuf, Nn, Nn, Nn, Nn, Nn, Nn, 0,
       Bb, 0, NN_, NN_, 0, 0, 0);
  k_softmax_n<<<gr(Bb * Nn), TPB, 0, stream>>>(Sbuf, SAt);

  // 3) Chebyshev graph conv
  k_tkt<<<gr(3LL * BNN), TPB, 0, stream>>>(chebP, SAt, TkT);
  // stage 1: Z[k,b] = TkT[k,b]^ @ trend[b]   (24 batches, z = k*8+b)
  gemm(stream, TkT, trend, nullptr, nullptr, Zb, Nn, Tt * Ff, Nn, Nn, Tt * Ff,
       Tt * Ff, 0, 3 * Bb, NN_, PANEL, PANEL, Bb, 0, 0);
  // stage 2: tes = relu(sum_k Z[k] @ theta[k])  (accumulating passes)
  for (int k = 0; k < 3; ++k)
    gemm(stream, Zb + (size_t)k * NEL, chebT + (size_t)k * Ff * Ff, nullptr,
         nullptr, tes, R, Ff, Ff, Ff, Ff, Ff, 0, 1, 0, 0, 0, 0, k > 0,
         k == 2 ? 1 : 0);

  // 4) bidirectional Mamba on xt = trend
  for (int d = 0; d < 2; ++d) {
    const float* xin = d ? gen1 : trend;
    if (d) k_flip<<<gr(NEL), TPB, 0, stream>>>(trend, gen1, Ff);
    k_rmsnorm<<<gr(R), TPB, 0, stream>>>(xin, mNormW + d * Ff, gen2);
    gemm(stream, gen2, mInProj + d * Ff * 256, nullptr, nullptr, big, R, 256, Ff,
         Ff, 256, 256, 0, 1, 0, 0, 0, 0, 0, 0);
    k_conv_silu<<<gr((long long)R * ED), TPB, 0, stream>>>(
        big, mConvW + d * ED * 4, mConvB + d * ED, xrb);
    gemm(stream, xrb, mXProj + d * ED * 36, nullptr, nullptr, dbc, R, 36, ED, ED,
         36, 36, 0, 1, 0, 0, 0, 0, 0, 0);
    k_scan<<<gr((long long)BN * ED), TPB, 0, stream>>>(
        dbc, mDtW + d * 4 * ED, mDtB + d * ED, mAlog + d * ED * DSs, mD + d * ED,
        big, xrb);
    if (d == 0) {
      gemm(stream, xrb, mOutP, nullptr, xin, cat, R, Ff, ED, ED, Ff, 128, Ff, 1,
           0, 0, 0, 0, 0, 0);
    } else {
      gemm(stream, xrb, mOutP + ED * Ff, nullptr, xin, gen2, R, Ff, ED, ED, Ff,
           Ff, Ff, 1, 0, 0, 0, 0, 0, 0);
      k_flip_to_cat<<<gr(NEL), TPB, 0, stream>>>(gen2, cat);
    }
  }
  // bim projection + LN -> trend_enc_t
  gemm(stream, cat, bimW, bimB, nullptr, gbuf, R, Ff, 128, 128, Ff, Ff, 0, 1, 0,
       0, 0, 0, 0, 1);
  k_layernorm<<<gr(R), TPB, 0, stream>>>(trend, gbuf, bimG, bimBe, tet);

  // 5) trend gate -> d_out (trend_enc)
  k_concat2<<<gr(NEL), TPB, 0, stream>>>(tes, tet, cat);
  gemm(stream, cat, tgW, tgB, nullptr, gbuf, R, Ff, 128, 128, Ff, Ff, 0, 1, 0, 0,
       0, 0, 0, 2);
  k_blend<<<gr(NEL), TPB, 0, stream>>>(gbuf, tes, tet, out);

  // 6) seasonal: graph attention over nodes
  k_to_btnf<<<gr(NEL), TPB, 0, stream>>>(seas, gen1);  // xs (B,T,N,F)
  gemm(stream, gen1, gQkvW, gQkvB, nullptr, qkvZ, R, 192, Ff, Ff, 192, 192, 0, 1,
       0, 0, 0, 0, 0, 0);
  k_attn<<<gr((long long)Bb * Tt * 8 * Nn), TPB, 0, stream>>>(qkvZ, adj, gen2,
                                                              Bb * Tt, Nn);
  gemm(stream, gen2, gOutW, gOutB, nullptr, gbuf, R, Ff, Ff, Ff, Ff, Ff, 0, 1, 0,
       0, 0, 0, 0, 0);
  k_layernorm<<<gr(R), TPB, 0, stream>>>(gen1, gbuf, gLn1g, gLn1b, gen1);
  gemm(stream, gen1, gFf1w, gFf1b, nullptr, gbuf, R, Ff, Ff, Ff, Ff, Ff, 0, 1, 0,
       0, 0, 0, 0, 1);
  gemm(stream, gbuf, gFf2w, gFf2b, nullptr, gen2, R, Ff, Ff, Ff, Ff, Ff, 0, 1, 0,
       0, 0, 0, 0, 0);
  k_layernorm<<<gr(R), TPB, 0, stream>>>(gen1, gen2, gLn2g, gLn2b, gen1);
  k_from_btnf<<<gr(NEL), TPB, 0, stream>>>(gen1, sesB);

  // 7) seasonal: temporal attention
  gemm(stream, seas, tQkvW, tQkvB, nullptr, qkvZ, R, 192, Ff, Ff, 192, 192, 0, 1,
       0, 0, 0, 0, 0, 0);
  k_attn<<<gr((long long)BN * 8 * Tt), TPB, 0, stream>>>(qkvZ, nullptr, gen1, BN,
                                                         Tt);
  gemm(stream, gen1, tOutW, tOutB, nullptr, gen2, R, Ff, Ff, Ff, Ff, Ff, 0, 1, 0,
       0, 0, 0, 0, 0);
  k_layernorm<<<gr(R), TPB, 0, stream>>>(seas, gen2, tLn1g, tLn1b, gen1);
  gemm(stream, gen1, tFf1w, tFf1b, nullptr, big, R, 256, Ff, Ff, 256, 256, 0, 1,
       0, 0, 0, 0, 0, 1);
  gemm(stream, big, tFf2w, tFf2b, nullptr, gen2, R, Ff, 256, 256, Ff, Ff, 0, 1, 0,
       0, 0, 0, 0, 0);
  k_layernorm<<<gr(R), TPB, 0, stream>>>(gen1, gen2, tLn2g, tLn2b, setB);

  // 8) seasonal gate + final residual sum
  k_concat2<<<gr(NEL), TPB, 0, stream>>>(sesB, setB, cat);
  gemm(stream, cat, sgW, sgB, nullptr, gbuf, R, Ff, 128, 128, Ff, Ff, 0, 1, 0, 0,
       0, 0, 0, 2);
  k_final<<<gr(NEL), TPB, 0, stream>>>(x, gbuf, sesB, setB, out);
}